// elr_loss_42150809043771
// MI455X (gfx1250) — compile-verified
//
#include <hip/hip_runtime.h>
#include <hip/hip_bf16.h>
#include <math.h>

typedef __attribute__((ext_vector_type(2))) float v2f;
typedef __attribute__((ext_vector_type(8))) float v8f;

#define BB   4096
#define CC   100
#define DD   512
#define NNN  1000000
#define LDW  104          // padded row stride for [B,C] buffers (16B-aligned rows)
#define EPSV 1e-4f
#define BETA 0.3f
#define LAMB 3.0f

// workspace layout (float offsets)
#define SC_OFF  0                          // scalars: 0=ce_sum 1=elr_sum 2=wnorm2 3=feat_sum
#define RN_OFF  16                         // rownorm2[4096]
#define ACC_OFF 4112                       // gemm2 accum [112*512]
#define ZERO_CNT 61456                     // SC+RN+ACC zeroed each call
#define W_OFF   61456                      // weight/t_read [4096*104]
#define YP_OFF  487440                     // clipped y_pred [4096*104]
#define VV_OFF  913424                     // v_vertical -> rot [4096*512]

__device__ __forceinline__ float waveRedSum(float v) {
#pragma unroll
  for (int m = 16; m >= 1; m >>= 1) v += __shfl_xor(v, m, 32);
  return v;
}

__global__ void k_init(float* ws) {
  int i = blockIdx.x * 256 + threadIdx.x;
  if (i < ZERO_CNT) ws[i] = 0.0f;
}

// One wave per row: softmax, clip, renorm, t_new, CE.
__global__ __launch_bounds__(256) void k_row(const float* __restrict__ logits,
                                             const int* __restrict__ index,
                                             const int* __restrict__ label,
                                             const float* __restrict__ target,
                                             float* ws) {
  int wave = threadIdx.x >> 5, lane = threadIdx.x & 31;
  int b = blockIdx.x * 8 + wave;                         // grid = 512 exact
  const float* o = logits + (size_t)b * CC;
  float vals[4]; int cols[4];
  float mx = -3.4e38f;
#pragma unroll
  for (int j = 0; j < 4; j++) {
    int c = lane + 32 * j; cols[j] = c;
    float v = (c < CC) ? o[c] : -3.4e38f;
    vals[j] = v; mx = fmaxf(mx, v);
  }
#pragma unroll
  for (int m = 16; m >= 1; m >>= 1) mx = fmaxf(mx, __shfl_xor(mx, m, 32));
  float z = 0.0f, e[4];
#pragma unroll
  for (int j = 0; j < 4; j++) { float v = (cols[j] < CC) ? __expf(vals[j] - mx) : 0.0f; e[j] = v; z += v; }
  z = waveRedSum(z);
  float invz = 1.0f / z;
  float s2 = 0.0f, yp[4];
#pragma unroll
  for (int j = 0; j < 4; j++) {
    float y = (cols[j] < CC) ? fminf(fmaxf(e[j] * invz, EPSV), 1.0f - EPSV) : 0.0f;
    yp[j] = y; s2 += y;
  }
  s2 = waveRedSum(s2);
  float invs2 = 1.0f / s2;
  int idx = index[b];
  const float* trow = target + (size_t)idx * CC;
  float* wrow  = ws + W_OFF  + (size_t)b * LDW;
  float* yprow = ws + YP_OFF + (size_t)b * LDW;
#pragma unroll
  for (int j = 0; j < 4; j++) {
    int c = cols[j];
    if (c < CC) {
      float p  = yp[j] * invs2;
      float tn = BETA * trow[c] + (1.0f - BETA) * p;
      wrow[c] = tn; yprow[c] = yp[j];
    } else if (c < LDW) { wrow[c] = 0.0f; yprow[c] = 0.0f; }
  }
  int lab = label[b];
  float logz = logf(z);
#pragma unroll
  for (int j = 0; j < 4; j++)
    if (cols[j] == lab) atomicAdd(ws + SC_OFF + 0, vals[j] - mx - logz);  // sum of logp[label]
}

__global__ void k_scatter(const int* __restrict__ index, const float* __restrict__ ws,
                          float* __restrict__ out_target) {
  int b = blockIdx.x, t = threadIdx.x;                   // 128 threads
  int idx = index[b];
  if (t < CC) out_target[(size_t)idx * CC + t] = ws[W_OFF + (size_t)b * LDW + t];
}

// Re-gather t_read from updated bank (duplicate-index faithful), ELR + ||weight||^2.
__global__ __launch_bounds__(256) void k_gather(const int* __restrict__ index,
                                                const float* __restrict__ out_target,
                                                float* ws) {
  int wave = threadIdx.x >> 5, lane = threadIdx.x & 31;
  int b = blockIdx.x * 8 + wave;
  int idx = index[b];
  const float* trow = out_target + (size_t)idx * CC;
  float* wrow = ws + W_OFF + (size_t)b * LDW;
  const float* yprow = ws + YP_OFF + (size_t)b * LDW;
  float dot = 0.0f, wn2 = 0.0f;
#pragma unroll
  for (int j = 0; j < 4; j++) {
    int c = lane + 32 * j;
    if (c < CC) {
      float tr = trow[c];
      wrow[c] = tr;
      dot += tr * yprow[c];
      wn2 += tr * tr;
    }
  }
  dot = waveRedSum(dot);
  wn2 = waveRedSum(wn2);
  if (lane == 0) {
    atomicAdd(ws + SC_OFF + 1, logf(1.0f - dot));
    atomicAdd(ws + SC_OFF + 2, wn2);
  }
}

// v_parallel = W[4096,100] @ U[100,512] via f32 WMMA; store v_vertical, row norms, feat loss.
__global__ __launch_bounds__(256) void k_gemm1(float* ws, const float* __restrict__ U,
                                               const float* __restrict__ vt) {
  int wave = threadIdx.x >> 5, lane = threadIdx.x & 31;
  int tile = blockIdx.x * 8 + wave;                      // 8192 tiles exact
  int tm = tile >> 5, tn = tile & 31;
  int mloc = lane & 15, nloc = lane & 15;
  int khalf = (lane >> 4) << 1;                          // ISA A layout: lanes 16-31 hold K+2
  const float* arow = ws + W_OFF + (size_t)(tm * 16 + mloc) * LDW + khalf;
  const float* bcol = U + (size_t)khalf * DD + tn * 16 + nloc;
  v8f acc = {};
  for (int k = 0; k < CC; k += 4) {
    v2f a, bm;
    a.x  = arow[k];  a.y  = arow[k + 1];
    bm.x = bcol[(size_t)k * DD]; bm.y = bcol[(size_t)(k + 1) * DD];
    acc = __builtin_amdgcn_wmma_f32_16x16x4_f32(false, a, false, bm, (short)0, acc,
                                                false, false);
  }
  float* vvert = ws + VV_OFF;
  int n = tn * 16 + nloc;
  float featpart = 0.0f;
#pragma unroll
  for (int r = 0; r < 8; r++) {
    int m = tm * 16 + r + ((lane >> 4) << 3);            // C/D layout: lane>=16 -> M+8
    float vv = vt[(size_t)m * DD + n] - acc[r];
    vvert[(size_t)m * DD + n] = vv;
    float ss = vv * vv;
    featpart += ss;
#pragma unroll
    for (int mm = 8; mm >= 1; mm >>= 1) ss += __shfl_xor(ss, mm, 32);  // within 16-lane group
    if (nloc == 0) atomicAdd(ws + RN_OFF + m, ss);
  }
  featpart = waveRedSum(featpart);
  if (lane == 0) atomicAdd(ws + SC_OFF + 3, featpart);
}

// rot (reproduces reference's v_parallel_norm := v_vertical_norm broadcast).
__global__ void k_rot(float* ws, const float* __restrict__ vt, const int* __restrict__ epoch) {
  size_t i = (size_t)blockIdx.x * 256 + threadIdx.x;
  if (i >= (size_t)BB * DD) return;
  int b = (int)(i / DD);
  float vv = ws[VV_OFF + i];
  float vp = vt[i] - vv;
  float r2 = ws[RN_OFF + b];
  float n_size = 1.0f / (float)epoch[0];
  float vvn = sqrtf(r2);
  float s = r2 * n_size;
  ws[VV_OFF + i] = ((__cosf(s) - 1.0f) * vp + __sinf(s) * vv) / vvn;
}

// acc[112,512] += W.T[100p,4096] @ rot[4096,512], K split in 8 chunks.
__global__ __launch_bounds__(256) void k_gemm2(float* ws) {
  int wave = threadIdx.x >> 5, lane = threadIdx.x & 31;
  int wid = blockIdx.x * 8 + wave;                       // 1792 waves exact
  int tm = wid >> 8;                                     // 0..6  (M padded to 112)
  int rem = wid & 255;
  int tn = rem >> 3;                                     // 0..31
  int kc = rem & 7;                                      // K chunk of 512
  int mloc = lane & 15, nloc = lane & 15;
  int khalf = (lane >> 4) << 1;
  int m = tm * 16 + mloc;
  bool mok = (m < CC);
  const float* acol = ws + W_OFF + m;                    // A[m][kb] = W[kb*LDW + m]
  const float* brow = ws + VV_OFF + (size_t)khalf * DD + tn * 16 + nloc;
  int k0 = kc * 512;
  v8f acc = {};
  for (int k = k0; k < k0 + 512; k += 4) {
    v2f a, bm;
    size_t kk = (size_t)(k + khalf);
    a.x  = mok ? acol[kk * LDW] : 0.0f;
    a.y  = mok ? acol[(kk + 1) * LDW] : 0.0f;
    bm.x = brow[(size_t)k * DD];
    bm.y = brow[(size_t)(k + 1) * DD];
    acc = __builtin_amdgcn_wmma_f32_16x16x4_f32(false, a, false, bm, (short)0, acc,
                                                false, false);
  }
  int n = tn * 16 + nloc;
#pragma unroll
  for (int r = 0; r < 8; r++) {
    int mm = tm * 16 + r + ((lane >> 4) << 3);
    atomicAdd(ws + ACC_OFF + (size_t)mm * DD + n, acc[r]);
  }
}

__global__ void k_final(const float* __restrict__ ws, const float* __restrict__ U,
                        float* __restrict__ out) {
  int i = blockIdx.x * 256 + threadIdx.x;
  if (i < CC * DD) {
    float wn = sqrtf(ws[SC_OFF + 2]);
    out[1 + (size_t)NNN * CC + i] = U[i] + ws[ACC_OFF + i] / wn;
  }
  if (blockIdx.x == 0 && threadIdx.x == 0) {
    float ce   = -ws[SC_OFF + 0] / (float)BB;
    float elr  =  ws[SC_OFF + 1] / (float)BB;
    float feat =  ws[SC_OFF + 3] / ((float)BB * (float)DD);
    out[0] = ce + LAMB * elr + feat;
  }
}

extern "C" void kernel_launch(void* const* d_in, const int* in_sizes, int n_in,
                              void* d_out, int out_size, void* d_ws, size_t ws_size,
                              hipStream_t stream) {
  const int*   index     = (const int*)d_in[0];
  const float* output    = (const float*)d_in[1];
  const int*   label     = (const int*)d_in[2];
  const float* vt        = (const float*)d_in[3];
  const int*   epoch     = (const int*)d_in[4];
  const float* target    = (const float*)d_in[5];
  const float* memory_ut = (const float*)d_in[6];
  float* out = (float*)d_out;
  float* ws  = (float*)d_ws;
  float* out_target = out + 1;

  // 1) bulk bank copy (the 400MB long pole; ~34us at 23.3 TB/s)
  hipMemcpyAsync(out_target, target, (size_t)NNN * CC * sizeof(float),
                 hipMemcpyDeviceToDevice, stream);
  // 2) zero atomic accumulators (ws is poisoned)
  k_init<<<(ZERO_CNT + 255) / 256, 256, 0, stream>>>(ws);
  // 3) per-row softmax / t_new / CE
  k_row<<<BB / 8, 256, 0, stream>>>(output, index, label, target, ws);
  // 4) scatter updated rows into output bank
  k_scatter<<<BB, 128, 0, stream>>>(index, ws, out_target);
  // 5) re-gather t_read; ELR + ||weight||^2
  k_gather<<<BB / 8, 256, 0, stream>>>(index, out_target, ws);
  // 6) GEMM1 (WMMA f32): v_parallel / v_vertical / norms / feature loss
  k_gemm1<<<(BB / 16) * (DD / 16) / 8, 256, 0, stream>>>(ws, memory_ut, vt);
  // 7) rot elementwise
  k_rot<<<((size_t)BB * DD + 255) / 256, 256, 0, stream>>>(ws, vt, epoch);
  // 8) GEMM2 (WMMA f32): weight.T @ rot, K-split + atomic accumulate
  k_gemm2<<<7 * 32 * 8 / 8, 256, 0, stream>>>(ws);
  // 9) finalize memory_ut_new + scalar loss
  k_final<<<(CC * DD + 255) / 256, 256, 0, stream>>>(ws, memory_ut, out);
}